// Wavelets_torch_36996848287852
// MI455X (gfx1250) — compile-verified
//
#include <hip/hip_runtime.h>

// CDNA5 / gfx1250 wave32 WMMA implementation of 4-level periodized db4
// wavedec2 + waverec2 on [8, 2048, 2048] f32.
//
// Each 1D periodized DWT pass is a banded GEMM on V_WMMA_F32_16X16X4_F32
// (full f32). Tiles are staged into LDS with GLOBAL_LOAD_ASYNC_TO_LDS_*
// (ASYNCcnt-tracked); strided-axis results are transposed through LDS so
// every global store is coalesced.

typedef float v2f __attribute__((ext_vector_type(2)));
typedef float v8f __attribute__((ext_vector_type(8)));
typedef int gv4i __attribute__((vector_size(4 * sizeof(int))));

#define PT 8          // perp 16-line tiles per wave
#define PITCH 2048    // row pitch of every buffer
#define IMGS (2048 * 2048)

#if defined(__has_builtin)
#if __has_builtin(__builtin_amdgcn_global_load_async_to_lds_b128) && \
    __has_builtin(__builtin_amdgcn_global_load_async_to_lds_b32) &&  \
    __has_builtin(__builtin_amdgcn_s_wait_asynccnt)
#define USE_ASYNC 1
#endif
#endif
#ifndef USE_ASYNC
#define USE_ASYNC 0
#endif

#if USE_ASYNC
// Builtin signature (from hipcc diagnostics): arg0 = AS1 (global) typed
// pointer, arg1 = AS3 (LDS) typed pointer, then imm offset + imm cpol.
#define G128(p) ((__attribute__((address_space(1))) gv4i*)(p))
#define L128(p) ((__attribute__((address_space(3))) gv4i*)(p))
#define G32(p) ((__attribute__((address_space(1))) int*)(p))
#define L32(p) ((__attribute__((address_space(3))) int*)(p))
__device__ __forceinline__ void async_wait0() {
  __builtin_amdgcn_s_wait_asynccnt(0);
}
#endif

// Pin a B fragment in VGPRs: blocks rematerialization of its select chain
// inside the tile loop (it must be built exactly once per wave).
__device__ __forceinline__ void pin(v2f& x) { asm volatile("" : "+v"(x)); }

// db4 analysis filters (hi = QMF of lo), baked as immediates.
template <int HIF>
__device__ __forceinline__ float tapval(int t) {
  constexpr float LOt[8] = {
      -0.010597401784997278f, 0.032883011666982945f, 0.030841381835986965f,
      -0.18703481171888114f,  -0.02798376941698385f, 0.6308807679295904f,
      0.7148465705525415f,    0.23037781330885523f};
  constexpr float HIt[8] = {
      0.23037781330885523f,  -0.7148465705525415f,  0.6308807679295904f,
      0.02798376941698385f,  -0.18703481171888114f, -0.030841381835986965f,
      0.032883011666982945f, 0.010597401784997278f};
  float v = 0.0f;
#pragma unroll
  for (int i = 0; i < 8; ++i) {
    float c = HIF ? HIt[i] : LOt[i];
    v = (t == i) ? c : v;
  }
  return v;
}

// ---------------------------------------------------------------------------
// Analysis along one axis: y_a[p,n] = sum_k lo[k] x[p,(2n+k) mod N] (d: hi).
// a -> along index [0,N/2), d -> [N/2,N).
// GEMM per tile: D[16 perp,16 out] = A[16,40] * B[40,16], 10 K=4 WMMA chunks,
// lo/hi accumulators share the A fragments.
// CONTIG=1: filtered axis unit stride (W pass), LDS tile [perp][along].
// CONTIG=0: H pass, LDS tile [along][perp] so 16B global runs stay contiguous.
// ---------------------------------------------------------------------------
template <int CONTIG>
__global__ __launch_bounds__(32) void dwt_analysis(
    const float* __restrict__ in, float* __restrict__ out, int N, int lgN,
    int totalPerp) {
  constexpr int sAlong = CONTIG ? 1 : PITCH;
  constexpr int sPerp = CONTIG ? PITCH : 1;
  __shared__ float lds[16 * 44];   // CONTIG: [16][44]; else [40][16]
  __shared__ float xsA[16 * 19];   // store-transpose buffers (H pass)
  __shared__ float xsD[16 * 19];
  const int lane = threadIdx.x;
  const int h = lane >> 4;         // lane half -> K base (ISA 16x4 f32 layout)
  const int nl = lane & 15;        // A: M index; B/D: N index
  const int n0 = blockIdx.x << 4;  // output subband column base
  const int jBase = n0 << 1;
  const int N2 = N >> 1;

  // B[k,n] = f[k-2n], per-lane K = 4q + v + 2h. Built once, pinned in VGPRs.
  v2f blo[10], bhi[10];
#pragma unroll
  for (int q = 0; q < 10; ++q) {
    int k0 = 4 * q + 2 * h;
    blo[q].x = tapval<0>(k0 - 2 * nl);
    blo[q].y = tapval<0>(k0 + 1 - 2 * nl);
    bhi[q].x = tapval<1>(k0 - 2 * nl);
    bhi[q].y = tapval<1>(k0 + 1 - 2 * nl);
    pin(blo[q]);
    pin(bhi[q]);
  }

#pragma unroll 1
  for (int t = 0; t < PT; ++t) {
    int pTile = (blockIdx.y * PT + t) << 4;
    if (pTile >= totalPerp) break;       // uniform
    int b = pTile >> lgN;                // tile never crosses an image (16|N)
    int r0 = pTile - (b << lgN);
    const float* inb = in + b * IMGS;
    float* outb = out + b * IMGS;

    __syncthreads();
    // Stage 16 perp x 40 along (all 16B-granular, quad-aligned, wrap-safe).
#pragma unroll
    for (int i = 0; i < 5; ++i) {
      int idx = lane + 32 * i;
      const float* src;
      float* dst;
      if (CONTIG) {  // 16 rows x 10 quads along the unit-stride axis
        int pp = idx / 10;
        int qi = idx - pp * 10;
        int j = jBase + 4 * qi;
        if (j >= N) j -= N;
        src = inb + (r0 + pp) * sPerp + j;
        dst = &lds[pp * 44 + 4 * qi];
      } else {       // 40 along-rows x 4 quads along the unit-stride perp axis
        int aa = idx >> 2;
        int ci = idx & 3;
        int j = jBase + aa;
        if (j >= N) j -= N;
        src = inb + j * sAlong + r0 + 4 * ci;
        dst = &lds[aa * 16 + 4 * ci];
      }
#if USE_ASYNC
      __builtin_amdgcn_global_load_async_to_lds_b128(G128(src), L128(dst), 0, 0);
#else
      *(float4*)dst = *(const float4*)src;
#endif
    }
#if USE_ASYNC
    async_wait0();
#endif
    __syncthreads();

    v8f accA = {};
    v8f accD = {};
#pragma unroll
    for (int q = 0; q < 10; ++q) {
      int k0 = 4 * q + 2 * h;
      v2f a;
      if (CONTIG) {
        a.x = lds[nl * 44 + k0];
        a.y = lds[nl * 44 + k0 + 1];
      } else {
        a.x = lds[k0 * 16 + nl];
        a.y = lds[(k0 + 1) * 16 + nl];
      }
      accA = __builtin_amdgcn_wmma_f32_16x16x4_f32(false, a, false, blo[q],
                                                   (short)0, accA, false, false);
      accD = __builtin_amdgcn_wmma_f32_16x16x4_f32(false, a, false, bhi[q],
                                                   (short)0, accD, false, false);
    }

    if (CONTIG) {
      // D: lane -> out col, VGPR v -> row v+8h. Rows are coalesced already.
      float* o = outb + (r0 + 8 * h) * sPerp + n0 + nl;
#pragma unroll
      for (int v = 0; v < 8; ++v) {
        o[v * sPerp] = accA[v];
        o[v * sPerp + N2] = accD[v];
      }
    } else {
      // Transpose through LDS so stores run along the unit-stride perp axis.
#pragma unroll
      for (int v = 0; v < 8; ++v) {
        xsA[nl * 19 + v + 8 * h] = accA[v];
        xsD[nl * 19 + v + 8 * h] = accD[v];
      }
      __syncthreads();
      float* o = outb + r0 + nl;  // sPerp == 1
#pragma unroll
      for (int w = 0; w < 8; ++w) {
        int nn = w + 8 * h;
        o[(n0 + nn) * sAlong] = xsA[nn * 19 + nl];
        o[(N2 + n0 + nn) * sAlong] = xsD[nn * 19 + nl];
      }
    }
  }
}

// ---------------------------------------------------------------------------
// Synthesis along one axis: x[p,j] = sum_n lo[j-2n] a[p,n] + hi[j-2n] d[p,n]
// (periodized). a/d interleaved in K (even K = a/lo, odd K = d/hi): K=24 band,
// 6 K=4 WMMA chunks. B-VGPR0 holds lo taps, B-VGPR1 hi taps.
// ---------------------------------------------------------------------------
template <int CONTIG>
__global__ __launch_bounds__(32) void dwt_synthesis(
    const float* __restrict__ in, float* __restrict__ out, int N, int lgN,
    int totalPerp) {
  constexpr int sAlong = CONTIG ? 1 : PITCH;
  constexpr int sPerp = CONTIG ? PITCH : 1;
  __shared__ float ldsA[16 * 14];  // [perp 16][coarse 12], stride 14
  __shared__ float ldsD[16 * 14];
  __shared__ float xs[16 * 19];    // store-transpose buffer (H pass)
  const int lane = threadIdx.x;
  const int h = lane >> 4;
  const int nl = lane & 15;
  const int j0 = blockIdx.x << 4;   // output along-axis base
  const int N2 = N >> 1;
  const int nbase = (j0 >> 1) - 3;  // first contributing coarse index

  // tap = (j0+nl) - 2*(nbase + (K>>1)) = nl + 6 - 2*(2q+h)
  v2f bq[6];
#pragma unroll
  for (int q = 0; q < 6; ++q) {
    int tap = nl + 6 - 2 * (2 * q + h);
    bq[q].x = tapval<0>(tap);  // even K -> a-subband -> lo
    bq[q].y = tapval<1>(tap);  // odd  K -> d-subband -> hi
    pin(bq[q]);
  }

#pragma unroll 1
  for (int t = 0; t < PT; ++t) {
    int pTile = (blockIdx.y * PT + t) << 4;
    if (pTile >= totalPerp) break;
    int b = pTile >> lgN;
    int r0 = pTile - (b << lgN);
    const float* inb = in + b * IMGS;
    float* outb = out + b * IMGS;

    __syncthreads();
    // Stage both subband planes: 2 x 16 perp x 12 coarse samples.
#pragma unroll
    for (int i = 0; i < 12; ++i) {
      int idx = lane + 32 * i;
      int s = idx >= 192;
      int rem = idx - s * 192;
      int pp, c;
      if (CONTIG) { pp = rem / 12; c = rem - pp * 12; }
      else        { pp = rem & 15; c = rem >> 4; }
      int nidx = nbase + c;
      if (nidx < 0) nidx += N2;
      else if (nidx >= N2) nidx -= N2;  // periodization
      const float* src = inb + (r0 + pp) * sPerp + (s * N2 + nidx) * sAlong;
      float* dst = (s ? ldsD : ldsA) + pp * 14 + c;
#if USE_ASYNC
      __builtin_amdgcn_global_load_async_to_lds_b32(G32(src), L32(dst), 0, 0);
#else
      *dst = *src;
#endif
    }
#if USE_ASYNC
    async_wait0();
#endif
    __syncthreads();

    v8f acc = {};
#pragma unroll
    for (int q = 0; q < 6; ++q) {
      int c = 2 * q + h;  // coarse index for this K pair
      v2f a;
      a.x = ldsA[nl * 14 + c];
      a.y = ldsD[nl * 14 + c];
      acc = __builtin_amdgcn_wmma_f32_16x16x4_f32(false, a, false, bq[q],
                                                  (short)0, acc, false, false);
    }

    if (CONTIG) {
      float* o = outb + (r0 + 8 * h) * sPerp + j0 + nl;
#pragma unroll
      for (int v = 0; v < 8; ++v) o[v * sPerp] = acc[v];
    } else {
#pragma unroll
      for (int v = 0; v < 8; ++v) xs[nl * 19 + v + 8 * h] = acc[v];
      __syncthreads();
      float* o = outb + r0 + nl;  // sPerp == 1
#pragma unroll
      for (int w = 0; w < 8; ++w) {
        int jj = w + 8 * h;
        o[(j0 + jj) * sAlong] = xs[jj * 19 + nl];
      }
    }
  }
}

extern "C" void kernel_launch(void* const* d_in, const int* in_sizes, int n_in,
                              void* d_out, int out_size, void* d_ws,
                              size_t ws_size, hipStream_t stream) {
  (void)in_sizes; (void)n_in; (void)out_size; (void)ws_size;
  const float* X = (const float*)d_in[0];
  float* C = (float*)d_out;  // coefficient pyramid; fully overwritten at end
  float* T = (float*)d_ws;   // ping buffer (128 MB), same pitched geometry

  // Forward: per level, analysis along W (contig) then along H (strided).
  for (int l = 0; l < 4; ++l) {
    int N = 2048 >> l;
    int lg = 11 - l;
    int totalPerp = 8 * N;  // 8 images x N active perp lines
    dim3 grid(N / 32, totalPerp / (16 * PT));
    dwt_analysis<1><<<grid, 32, 0, stream>>>((l == 0) ? X : C, T, N, lg,
                                             totalPerp);
    dwt_analysis<0><<<grid, 32, 0, stream>>>(T, C, N, lg, totalPerp);
  }

  // Inverse: per level, synthesis along H then along W (mirrors forward).
  for (int l = 3; l >= 0; --l) {
    int N = 2048 >> l;
    int lg = 11 - l;
    int totalPerp = 8 * N;
    dim3 grid(N / 16, totalPerp / (16 * PT));
    dwt_synthesis<0><<<grid, 32, 0, stream>>>(C, T, N, lg, totalPerp);
    dwt_synthesis<1><<<grid, 32, 0, stream>>>(T, C, N, lg, totalPerp);
  }
}